// TTFC_37374805410122
// MI455X (gfx1250) — compile-verified
//
#include <hip/hip_runtime.h>
#include <hip/hip_bf16.h>

#define C_DIM 120
#define B_DIM 4096
#define NP (C_DIM * B_DIM)     // 491520 (c,b) pairs
#define NCHUNK (NP / 4)        // 122880 chunks of 4 pairs
#define BLOCKS 2048
#define TPB 128                // 4 waves per block
#define WAVES_PER_BLOCK (TPB / 32)

typedef float v2f __attribute__((ext_vector_type(2)));
typedef float v8f __attribute__((ext_vector_type(8)));
typedef int v4i __attribute__((ext_vector_type(4)));

#define AS1 __attribute__((address_space(1)))
#define AS3 __attribute__((address_space(3)))

#if __has_builtin(__builtin_amdgcn_global_load_async_to_lds_b128)
#define HAS_ASYNC 1
#else
#define HAS_ASYNC 0
#endif

#if __has_builtin(__builtin_amdgcn_wmma_f32_16x16x4_f32)
#define HAS_WMMA_F32X4 1
#else
#define HAS_WMMA_F32X4 0
#endif

#if HAS_ASYNC
// (global v4i* src, local v4i* dst, imm byte offset applied to BOTH sides, cpol)
#define ASYNC_B128(GP, LP, OFF)                                               \
  __builtin_amdgcn_global_load_async_to_lds_b128((AS1 v4i*)(GP),              \
                                                 (AS3 v4i*)(LP), (OFF), 0)
#else
// synchronous fallback: still coalesced b128 load + ds_store
#define ASYNC_B128(GP, LP, OFF)                                               \
  do {                                                                        \
    *(float4*)((char*)(LP) + (OFF)) =                                         \
        *(const float4*)((const char*)(GP) + (OFF));                          \
  } while (0)
#endif

__device__ __forceinline__ void wait_async(bool more_pending) {
#if HAS_ASYNC
#if __has_builtin(__builtin_amdgcn_s_wait_asynccnt)
  if (more_pending)
    __builtin_amdgcn_s_wait_asynccnt(9);
  else
    __builtin_amdgcn_s_wait_asynccnt(0);
#else
  if (more_pending)
    asm volatile("s_wait_asynccnt 9" ::: "memory");
  else
    asm volatile("s_wait_asynccnt 0" ::: "memory");
#endif
#endif
}

#if HAS_WMMA_F32X4
__device__ __forceinline__ v8f wmma4(v2f a, v2f b, v8f c) {
  // D = A(16x4) * B(4x16) + C, all f32, exact
  return __builtin_amdgcn_wmma_f32_16x16x4_f32(false, a, false, b, (short)0, c,
                                               false, false);
}
#endif

// Stage one chunk (4 pairs) into this wave's LDS slab:
//   [0,2048)   : g=0  X data, 4 pairs x 512B
//   [2048,4096): g=1  X data
//   [4096,4608): core0 X data, 4 pairs x 128B
__device__ __forceinline__ void prefetch_chunk(float* wb, const float* x12,
                                               const float* x0, int ch,
                                               int lane) {
  const char* g0 = (const char*)x12 + (size_t)ch * 2048 + lane * 16;
  const char* g1 =
      (const char*)x12 + (size_t)NP * 512 + (size_t)ch * 2048 + lane * 16;
  const char* c0 = (const char*)x0 + (size_t)ch * 512 + lane * 16;
  char* l = (char*)wb + lane * 16;
  ASYNC_B128(g0, l, 0);
  ASYNC_B128(g0, l, 512);
  ASYNC_B128(g0, l, 1024);
  ASYNC_B128(g0, l, 1536);
  ASYNC_B128(g1, l + 2048, 0);
  ASYNC_B128(g1, l + 2048, 512);
  ASYNC_B128(g1, l + 2048, 1024);
  ASYNC_B128(g1, l + 2048, 1536);
  ASYNC_B128(c0, l + 4096, 0);
}

__global__ __launch_bounds__(TPB) void fused_antisym_norm_kernel(
    const float* __restrict__ x0,   // [C,B,1,8,4]
    const float* __restrict__ x12,  // [2,C,B,4,8,4]
    const float* __restrict__ w0,   // [1,8,4]
    const float* __restrict__ w12,  // [2,4,8,4]
    float* __restrict__ out) {      // [B,C]
  __shared__ __align__(16) float smem[WAVES_PER_BLOCK][2][1152];

  const int lane = threadIdx.x & 31;
  const int wid = threadIdx.x >> 5;
  const int wgid = blockIdx.x * WAVES_PER_BLOCK + wid;
  const int nw = gridDim.x * WAVES_PER_BLOCK;

  // Lane roles for the 16x16x4 f32 WMMA operand/result layouts.
  const int idx16 = lane & 15;
  const int half = lane >> 4;          // K-pair half: j{0,1} vs j{2,3}
  const int jb = half * 2;             // j base for this lane's 2 VGPRs
  const int ss = idx16 >> 2;           // s-sub (0..3) of this lane's row/col
  const int e = idx16 & 3;             // k / i / a component
  const bool act = (half == (ss >> 1));// lane holds diagonal-block rows?
  const int rsel = ss & 1;             // which VGPR quad holds them
  const int lane_xoff = jb * 32 + idx16;

  // Per-lane W operands (serve as both A and B side; identical f32 layout).
  v2f wop[2][2];
#pragma unroll
  for (int g = 0; g < 2; ++g)
#pragma unroll
    for (int h = 0; h < 2; ++h) {
      int s = h * 4 + ss;
      wop[g][h][0] = w12[g * 128 + jb * 32 + s * 4 + e];
      wop[g][h][1] = w12[g * 128 + (jb + 1) * 32 + s * 4 + e];
    }

  // core0: lane -> (s = lane>>2, comp = lane&3); W2 = |w_s|^2 per quad
  float w0v = w0[lane];
  float W2 = w0v * w0v;
  W2 += __shfl_xor(W2, 1, 32);
  W2 += __shfl_xor(W2, 2, 32);

  // eps folding: 24 slices * 16 * eps^2 + outer EPS
  const float CONST_TERM = 1.0e-5f + 384.0f * 1.0e-10f;

  int ch = wgid;
  int bufi = 0;
  if (ch < NCHUNK) prefetch_chunk(&smem[wid][0][0], x12, x0, ch, lane);

  for (; ch < NCHUNK; ch += nw) {
    int nxt = ch + nw;
    if (nxt < NCHUNK)
      prefetch_chunk(&smem[wid][bufi ^ 1][0], x12, x0, nxt, lane);
    wait_async(nxt < NCHUNK);  // current chunk's 9 async ops complete
    const float* wb = &smem[wid][bufi][0];

#pragma unroll
    for (int pp = 0; pp < 4; ++pp) {
      float part12 = 0.0f;
#pragma unroll
      for (int g = 0; g < 2; ++g) {
#pragma unroll
        for (int h = 0; h < 2; ++h) {
          float pick;
#if HAS_WMMA_F32X4
          int xo = g * 512 + pp * 128 + lane_xoff + h * 16;
          float xv0 = wb[xo];
          float xv1 = wb[xo + 32];
          v2f xb;  xb[0] = xv0;  xb[1] = xv1;
          v2f xn;  xn[0] = -xv0; xn[1] = -xv1;
          v8f cz = {0.f, 0.f, 0.f, 0.f, 0.f, 0.f, 0.f, 0.f};
          v8f d1 = wmma4(wop[g][h], xb, cz);       // diag blocks = M_s
          v8f p = wmma4(xn, wop[g][h], d1);        // = M_s - M_s^T (diag)
          float slo = fmaf(p[0], p[0],
                      fmaf(p[1], p[1], fmaf(p[2], p[2], p[3] * p[3])));
          float shi = fmaf(p[4], p[4],
                      fmaf(p[5], p[5], fmaf(p[6], p[6], p[7] * p[7])));
          pick = rsel ? shi : slo;
#else
          const float* Wg = w12 + g * 128;
          const float* Xs = wb + g * 512 + pp * 128;
          int s = h * 4 + ss;
          pick = 0.0f;
#pragma unroll
          for (int a = 0; a < 4; ++a) {
            float mab = 0.f, mba = 0.f;
#pragma unroll
            for (int j = 0; j < 4; ++j) {
              mab = fmaf(Wg[j * 32 + s * 4 + a], Xs[j * 32 + s * 4 + e], mab);
              mba = fmaf(Wg[j * 32 + s * 4 + e], Xs[j * 32 + s * 4 + a], mba);
            }
            float d = mab - mba;
            pick = fmaf(d, d, pick);
          }
#endif
          part12 += act ? pick : 0.0f;
        }
      }

      // core0 (rank-1): 0.5*(|w|^2 |x|^2 - (w.x)^2) per s, quad-reduced
      float xv = wb[1024 + pp * 32 + lane];
      float dv = w0v * xv;
      float x2 = xv * xv;
      dv += __shfl_xor(dv, 1, 32);
      dv += __shfl_xor(dv, 2, 32);
      x2 += __shfl_xor(x2, 1, 32);
      x2 += __shfl_xor(x2, 2, 32);
      // each quad lane adds 1/4 of its slice's 0.5*(...)
      float tot = fmaf(0.25f, part12, 0.125f * fmaf(W2, x2, -(dv * dv)));

      tot += __shfl_xor(tot, 16, 32);
      tot += __shfl_xor(tot, 8, 32);
      tot += __shfl_xor(tot, 4, 32);
      tot += __shfl_xor(tot, 2, 32);
      tot += __shfl_xor(tot, 1, 32);

      if (lane == 0) {
        int p = ch * 4 + pp;
        int c = p >> 12;       // / 4096
        int b = p & 4095;      // % 4096
        out[b * C_DIM + c] = sqrtf(tot + CONST_TERM);
      }
    }
    bufi ^= 1;
  }
}

extern "C" void kernel_launch(void* const* d_in, const int* in_sizes, int n_in,
                              void* d_out, int out_size, void* d_ws,
                              size_t ws_size, hipStream_t stream) {
  (void)in_sizes; (void)n_in; (void)out_size; (void)d_ws; (void)ws_size;
  const float* x0 = (const float*)d_in[0];
  const float* x12 = (const float*)d_in[1];
  const float* w0 = (const float*)d_in[2];
  const float* w12 = (const float*)d_in[3];
  float* out = (float*)d_out;
  fused_antisym_norm_kernel<<<dim3(BLOCKS), dim3(TPB), 0, stream>>>(
      x0, x12, w0, w12, out);
}